// TrainableKoopmanDynamics_85985245266124
// MI455X (gfx1250) — compile-verified
//
#include <hip/hip_runtime.h>
#include <cstdint>

// Koopman block-diagonal dynamics: elementwise streaming op, memory-bound.
// MI455X plan: async global->LDS staging (CDNA5 data-mover path), wave32
// row-parallel compute, fully coalesced b128 loads / b64 stores.

#define KDT 0.01f
#define KDIM 128          // 2*N_BLOCKS + M_REAL = 64 + 64
#define NBLK 32           // N_BLOCKS (complex pairs per row)
#define ROWS_PER_TILE 32  // rows staged per block -> 2 * 16 KB LDS
#define THREADS 256       // 8 wave32

__device__ __forceinline__ unsigned lds_off32(const void* p) {
    // Generic pointers into the LDS aperture carry the wave-relative LDS
    // byte offset in their low 32 bits (CDNA5 ISA 10.2 aperture mapping).
    return (unsigned)(uintptr_t)p;
}

__device__ __forceinline__ void async_copy_b128(unsigned lds_byte_off, uint64_t gaddr) {
    asm volatile("global_load_async_to_lds_b128 %0, %1, off"
                 :
                 : "v"(lds_byte_off), "v"(gaddr)
                 : "memory");
}

__global__ __launch_bounds__(THREADS)
void koopman_dynamics_kernel(const float* __restrict__ x,
                             const float* __restrict__ Lam,
                             float* __restrict__ out,
                             int batch) {
    __shared__ float sh_x[ROWS_PER_TILE * KDIM];  // 16 KB
    __shared__ float sh_L[ROWS_PER_TILE * KDIM];  // 16 KB

    const int row0 = blockIdx.x * ROWS_PER_TILE;
    int rows = batch - row0;
    if (rows <= 0) return;
    if (rows > ROWS_PER_TILE) rows = ROWS_PER_TILE;

    const int tid = threadIdx.x;

    // ---- Stage tile (x rows + Lambda rows) into LDS via async data mover ----
    {
        const unsigned lds_x = lds_off32(sh_x);
        const unsigned lds_L = lds_off32(sh_L);
        const uint64_t gx = (uint64_t)(x + (size_t)row0 * KDIM);
        const uint64_t gL = (uint64_t)(Lam + (size_t)row0 * KDIM);
        const int chunks = rows * (KDIM * 4 / 16);  // b128 chunks per array
        for (int i = tid; i < chunks; i += THREADS) {
            const unsigned off = (unsigned)i * 16u;
            async_copy_b128(lds_x + off, gx + off);
            async_copy_b128(lds_L + off, gL + off);
        }
        asm volatile("s_wait_asynccnt 0x0" ::: "memory");
    }
    __syncthreads();

    // ---- Compute: one wave32 per row; lane l owns pair l + reals 2l,2l+1 ----
    const int wave = tid >> 5;
    const int lane = tid & 31;

    for (int r = wave; r < rows; r += (THREADS / 32)) {
        const float* __restrict__ srx = sh_x + r * KDIM;
        const float* __restrict__ srL = sh_L + r * KDIM;
        float* __restrict__ orow = out + (size_t)(row0 + r) * KDIM;

        // Complex block part: y = e^{mu dt} * R(omega dt) * [x0, x1]
        const float x0 = srx[2 * lane + 0];
        const float x1 = srx[2 * lane + 1];
        const float mu = srL[lane];
        const float om = srL[NBLK + lane];
        const float e  = expf(mu * KDT);
        const float c  = cosf(om * KDT);
        const float s  = sinf(om * KDT);
        float2 yc;
        yc.x = e * (c * x0 - s * x1);
        yc.y = e * (s * x0 + c * x1);
        reinterpret_cast<float2*>(orow)[lane] = yc;

        // Real part: y = e^{lam dt} * x
        const float xr0 = srx[2 * NBLK + 2 * lane + 0];
        const float xr1 = srx[2 * NBLK + 2 * lane + 1];
        const float l0  = srL[2 * NBLK + 2 * lane + 0];
        const float l1  = srL[2 * NBLK + 2 * lane + 1];
        float2 yr;
        yr.x = expf(l0 * KDT) * xr0;
        yr.y = expf(l1 * KDT) * xr1;
        reinterpret_cast<float2*>(orow + 2 * NBLK)[lane] = yr;
    }
}

extern "C" void kernel_launch(void* const* d_in, const int* in_sizes, int n_in,
                              void* d_out, int out_size, void* d_ws, size_t ws_size,
                              hipStream_t stream) {
    (void)n_in; (void)d_ws; (void)ws_size; (void)out_size;
    const float* x   = (const float*)d_in[0];
    const float* Lam = (const float*)d_in[1];
    float* out = (float*)d_out;

    const int batch  = in_sizes[0] / KDIM;
    const int blocks = (batch + ROWS_PER_TILE - 1) / ROWS_PER_TILE;
    koopman_dynamics_kernel<<<blocks, THREADS, 0, stream>>>(x, Lam, out, batch);
}